// LightGCN_66090956751127
// MI455X (gfx1250) — compile-verified
//
#include <hip/hip_runtime.h>
#include <hip/hip_bf16.h>

// LightGCN propagation on MI455X (gfx1250).
// Memory/atomic-bound scatter-gather: per layer 5M edges * (256B gather +
// 256B scatter-add). Both the 76.8MB ego buffer and the 76.8MB destination
// fit in the 192MB L2, so gathers + f32 atomics are L2-resident after warmup.
// CDNA5 path: async global->LDS gather engine (per-lane addresses), 4-deep
// pipelined per wave (ASYNCcnt), LDS read-back via explicit ds_load_b64,
// scatter via explicit no-return global_atomic_add_f32.

#define D 64           // embedding dim
#define DEPTH 4        // async pipeline depth (LDS slots per wave)

// One edge per wave. Lane l owns columns [2l, 2l+1] of the 64-wide row.
// Source row for layer 1 is concat(user_emb, item_emb): col < nu -> srcA,
// else srcB at (col - nu). Later layers pass nu = N so srcA always wins.
// Each wave keeps up to DEPTH-1 async gathers in flight; slot i is consumed
// while slots i+1..i+3 are being filled by the async engine.
__global__ __launch_bounds__(256) void lgcn_spmm_kernel(
    const float* __restrict__ srcA, const float* __restrict__ srcB, int nu,
    float* __restrict__ dst,
    const float* __restrict__ vals, const int* __restrict__ rows,
    const int* __restrict__ cols, int nnz) {
    __shared__ __align__(16) float smem[8 * DEPTH * D];  // 8 waves * 4 slots * 256B = 8KB

    const int lane = threadIdx.x & 31;
    const int wib = threadIdx.x >> 5;                      // wave in block
    const int waveId = blockIdx.x * (blockDim.x >> 5) + wib;
    const int nWaves = gridDim.x * (blockDim.x >> 5);

    // LDS byte offset of smem[0]: generic LDS pointers carry the wave-relative
    // LDS address in their low 32 bits (ISA 10.2 aperture rules). Deriving the
    // asm operands from this also makes smem "escape", so the "memory" clobber
    // covers the async engine's LDS writes.
    const unsigned smemBase = (unsigned)(unsigned long long)(void*)smem;
    const unsigned myBase = smemBase + (unsigned)wib * (DEPTH * D * 4);

    auto issue = [&](int edge, int slot_) {
        int col = cols[edge];
        const float* s = (col < nu) ? (srcA + (size_t)col * D)
                                    : (srcB + (size_t)(col - nu) * D);
        unsigned loff = myBase + (unsigned)slot_ * (D * 4) + (unsigned)lane * 8u;
        unsigned long long ga = (unsigned long long)(s + lane * 2);
        asm volatile("global_load_async_to_lds_b64 %0, %1, off"
                     :: "v"(loff), "v"(ga) : "memory");
    };

    int e = waveId;
    if (e >= nnz) return;

    // Prime the pipeline with up to 3 gathers.
    issue(e, 0);
    if (e + nWaves < nnz)     issue(e + nWaves, 1);
    if (e + 2 * nWaves < nnz) issue(e + 2 * nWaves, 2);

    int slot = 0;
    for (; e < nnz; e += nWaves) {
        int ne3 = e + 3 * nWaves;  // max ~5.2M, fits in int
        if (ne3 < nnz) {
            issue(ne3, (slot + 3) & 3);
            // <=3 outstanding after this wait => the oldest (current slot) is done.
            asm volatile("s_wait_asynccnt 3" ::: "memory");
        } else {
            asm volatile("s_wait_asynccnt 0" ::: "memory");
        }

        float v = vals[e];
        int row = rows[e];

        // Explicit LDS read of this wave's slot; data is in VGPRs before the
        // asm block ends, so later async writes to this slot cannot race it.
        unsigned laddr = myBase + (unsigned)slot * (D * 4) + (unsigned)lane * 8u;
        unsigned long long xr;
        asm volatile("ds_load_b64 %0, %1\n\t"
                     "s_wait_dscnt 0"
                     : "=v"(xr) : "v"(laddr) : "memory");
        float m0 = v * __uint_as_float((unsigned)xr);
        float m1 = v * __uint_as_float((unsigned)(xr >> 32));

        // Explicit no-return f32 atomic adds (STOREcnt; ack at L2 scope where
        // dst is resident). s_endpgm's implicit wait-idle covers completion.
        unsigned long long daddr =
            (unsigned long long)(dst + (size_t)row * D + lane * 2);
        asm volatile("global_atomic_add_f32 %0, %1, off\n\t"
                     "global_atomic_add_f32 %0, %2, off offset:4"
                     :: "v"(daddr), "v"(m0), "v"(m1) : "memory");

        slot = (slot + 1) & 3;
    }
}

__global__ __launch_bounds__(256) void lgcn_zero2_kernel(float4* __restrict__ a,
                                                         float4* __restrict__ b,
                                                         long n4) {
    long i = (long)blockIdx.x * blockDim.x + threadIdx.x;
    long stride = (long)gridDim.x * blockDim.x;
    float4 z = make_float4(0.f, 0.f, 0.f, 0.f);
    for (; i < n4; i += stride) { a[i] = z; b[i] = z; }
}

// acc += src * scale; optionally zero zbuf (next layer's scatter target),
// fused so the dense pass streams each buffer exactly once.
__global__ __launch_bounds__(256) void lgcn_axpy_zero_kernel(float4* __restrict__ acc,
                                                             const float4* __restrict__ src,
                                                             float4* __restrict__ zbuf,
                                                             long n4, float scale) {
    long i = (long)blockIdx.x * blockDim.x + threadIdx.x;
    long stride = (long)gridDim.x * blockDim.x;
    float4 z = make_float4(0.f, 0.f, 0.f, 0.f);
    for (; i < n4; i += stride) {
        float4 s = src[i];
        float4 a = acc[i];
        a.x += s.x * scale; a.y += s.y * scale;
        a.z += s.z * scale; a.w += s.w * scale;
        acc[i] = a;
        if (zbuf) zbuf[i] = z;
    }
}

extern "C" void kernel_launch(void* const* d_in, const int* in_sizes, int n_in,
                              void* d_out, int out_size, void* d_ws, size_t ws_size,
                              hipStream_t stream) {
    const float* user_emb = (const float*)d_in[0];
    const float* item_emb = (const float*)d_in[1];
    const float* vals     = (const float*)d_in[2];
    const int*   rows     = (const int*)d_in[3];
    const int*   cols     = (const int*)d_in[4];

    const int nu  = in_sizes[0] / D;
    const int ni  = in_sizes[1] / D;
    const int nnz = in_sizes[2];
    const long N  = (long)nu + (long)ni;
    const long n4 = N * (D / 4);        // float4 elements per dense buffer

    float* acc  = (float*)d_out;        // running-mean accumulator [N, 64]
    float* ego0 = (float*)d_ws;         // ping
    float* ego1 = ego0 + N * D;         // pong

    const float third = 1.0f / 3.0f;
    const int DENSE_BLOCKS = 4096;
    const int SPMM_BLOCKS  = 8192;      // 65536 waves; ~76 edges per wave

    // acc = 0, ego0 = 0 (layer-1 scatter target)
    lgcn_zero2_kernel<<<DENSE_BLOCKS, 256, 0, stream>>>((float4*)acc, (float4*)ego0, n4);

    // Layer 1: ego0 = A * concat(user_emb, item_emb)
    lgcn_spmm_kernel<<<SPMM_BLOCKS, 256, 0, stream>>>(user_emb, item_emb, nu, ego0,
                                                      vals, rows, cols, nnz);
    // acc += ego0/3 ; zero ego1 for layer 2
    lgcn_axpy_zero_kernel<<<DENSE_BLOCKS, 256, 0, stream>>>((float4*)acc, (const float4*)ego0,
                                                            (float4*)ego1, n4, third);

    // Layer 2: ego1 = A * ego0   (nu = N -> single source pointer)
    lgcn_spmm_kernel<<<SPMM_BLOCKS, 256, 0, stream>>>(ego0, ego0, (int)N, ego1,
                                                      vals, rows, cols, nnz);
    // acc += ego1/3 ; zero ego0 for layer 3
    lgcn_axpy_zero_kernel<<<DENSE_BLOCKS, 256, 0, stream>>>((float4*)acc, (const float4*)ego1,
                                                            (float4*)ego0, n4, third);

    // Layer 3: ego0 = A * ego1
    lgcn_spmm_kernel<<<SPMM_BLOCKS, 256, 0, stream>>>(ego1, ego1, (int)N, ego0,
                                                      vals, rows, cols, nnz);
    // acc += ego0/3 ; done
    lgcn_axpy_zero_kernel<<<DENSE_BLOCKS, 256, 0, stream>>>((float4*)acc, (const float4*)ego0,
                                                            (float4*)nullptr, n4, third);

    (void)n_in; (void)out_size; (void)ws_size;
}